// Net_60224031424524
// MI455X (gfx1250) — compile-verified
//
#include <hip/hip_runtime.h>
#include <hip/hip_bf16.h>

typedef __attribute__((ext_vector_type(2))) float v2f;
typedef __attribute__((ext_vector_type(8))) float v8f;

// ---------------------------------------------------------------------------
// Zero-fill (graph-capture-safe replacement for hipMemsetAsync)
// ---------------------------------------------------------------------------
__global__ void zero_kernel(float* __restrict__ p, long n) {
    long i = (long)blockIdx.x * blockDim.x + threadIdx.x;
    long stride = (long)gridDim.x * blockDim.x;
    for (; i < n; i += stride) p[i] = 0.0f;
}

// ---------------------------------------------------------------------------
// Degree histogram: deg_out[src[e]] += 1, deg_in[dst[e]] += 1
// ---------------------------------------------------------------------------
__global__ void deg_kernel(const int* __restrict__ src, const int* __restrict__ dst,
                           float* __restrict__ deg_out, float* __restrict__ deg_in, int E) {
    int e = blockIdx.x * blockDim.x + threadIdx.x;
    if (e < E) {
        atomicAdd(&deg_out[src[e]], 1.0f);
        atomicAdd(&deg_in[dst[e]], 1.0f);
    }
}

// In-place deg -> rsqrt(max(deg,1))
__global__ void rsqrt_kernel(float* __restrict__ d, int n) {
    int i = blockIdx.x * blockDim.x + threadIdx.x;
    if (i < n) d[i] = rsqrtf(fmaxf(d[i], 1.0f));
}

// ---------------------------------------------------------------------------
// Generic fp32 WMMA GEMM (V_WMMA_F32_16X16X4_F32):
//   C[nrows x ncols] = diag(scaleA) * A[nrows x K] @ B[K x ncols] (+ bias)
//
// Register tiling: 256 threads = 8 waves = 4 row-tiles x 2 col-groups.
// Each wave computes 16 rows x 4 col-tiles (64 cols) with 4 accumulators, so
// one A fragment (b64 load) feeds 4 WMMAs. Block tile = 64 rows x 128 cols.
//
// Codegen points:
//  - A rows / B cols CLAMPED, not guarded: out-of-range rows/cols only corrupt
//    C entries the (fully predicated) store never writes -> zero bounds checks
//    and no exec-mask diamonds in the K loop.
//  - per-row scale hoisted to the store: diag(s)*A@B == s[m]*(A@B)[m,:].
//  - main loop is explicitly two-phase: ALL 18 loads (2x A b64 + 16x B b32)
//    into distinct locals first, THEN 8 WMMAs -> loads batch under staggered
//    loadcnt waits instead of load/wait-0/wmma serialization.
//  - EXEC all-1s at every WMMA (only uniform per-wave early exit).
// ---------------------------------------------------------------------------
__global__ __launch_bounds__(256)
void wmma_gemm_f32(const float* __restrict__ A, int lda,
                   const float* __restrict__ scaleA,
                   const float* __restrict__ B, int ldb, int K,
                   float* __restrict__ C, int ldc, int ncols,
                   const float* __restrict__ bias, int nrows) {
    const int lane = threadIdx.x & 31;   // wave32
    const int wave = threadIdx.x >> 5;   // 0..7
    const int half = lane >> 4;          // 0 or 1 (lane group)
    const int sub  = lane & 15;

    const int rowtile = wave >> 1;       // 0..3
    const int colgrp  = wave & 1;        // 0..1

    const int rowbase  = blockIdx.x * 64 + rowtile * 16;
    const int colbase0 = (blockIdx.y * 8 + colgrp * 4) * 16;  // first of 4 col tiles
    if (colbase0 >= ncols) return;       // uniform per-wave exit

    // A fragment: lanes 0-15 / 16-31 both hold M=sub; VGPR0=K+0/K+2, VGPR1=K+1/K+3
    const int rowA = min(rowbase + sub, nrows - 1);           // clamp, no guard
    const float* __restrict__ Arow = A + (long)rowA * lda;

    // 4 B column pointers (row K striped across lanes), clamped
    const float* __restrict__ Bc[4];
#pragma unroll
    for (int t = 0; t < 4; ++t)
        Bc[t] = B + min(colbase0 + t * 16 + sub, ncols - 1);

    const int kA = 2 * half;
    const int K4 = K & ~3;
    v8f acc[4] = {v8f{}, v8f{}, v8f{}, v8f{}};

    int kk = 0;
    // ---- main loop: 8 K per iteration, two-phase (loads then WMMAs) ----
    for (; kk + 8 <= K4; kk += 8) {
        __builtin_prefetch(Arow + kk + 32, 0, 3);

        const int k0 = kk + kA;          // even -> 8B aligned
        const int k1 = kk + 4 + kA;

        // ---- load phase: 2 A b64 + 16 B b32, all into distinct locals ----
        const v2f a0 = *(const v2f*)(Arow + k0);
        const v2f a1 = *(const v2f*)(Arow + k1);
        v2f b0[4], b1[4];
#pragma unroll
        for (int t = 0; t < 4; ++t) {
            b0[t].x = Bc[t][(long)k0 * ldb];
            b0[t].y = Bc[t][(long)(k0 + 1) * ldb];
            b1[t].x = Bc[t][(long)k1 * ldb];
            b1[t].y = Bc[t][(long)(k1 + 1) * ldb];
        }

        // ---- compute phase: 8 back-to-back WMMAs ----
#pragma unroll
        for (int t = 0; t < 4; ++t)
            acc[t] = __builtin_amdgcn_wmma_f32_16x16x4_f32(false, a0, false, b0[t],
                                                           (short)0, acc[t], false, false);
#pragma unroll
        for (int t = 0; t < 4; ++t)
            acc[t] = __builtin_amdgcn_wmma_f32_16x16x4_f32(false, a1, false, b1[t],
                                                           (short)0, acc[t], false, false);
    }
    // remaining whole K-steps
    for (; kk < K4; kk += 4) {
        const int k0 = kk + kA;
        const v2f a = *(const v2f*)(Arow + k0);
        v2f b[4];
#pragma unroll
        for (int t = 0; t < 4; ++t) {
            b[t].x = Bc[t][(long)k0 * ldb];
            b[t].y = Bc[t][(long)(k0 + 1) * ldb];
        }
#pragma unroll
        for (int t = 0; t < 4; ++t)
            acc[t] = __builtin_amdgcn_wmma_f32_16x16x4_f32(false, a, false, b[t],
                                                           (short)0, acc[t], false, false);
    }
    // tail (K % 4 != 0): only place with K guards, runs at most once
    if (kk < K) {
        const int k0 = kk + kA;
        v2f a;
        a.x = (k0     < K) ? Arow[k0]     : 0.0f;
        a.y = (k0 + 1 < K) ? Arow[k0 + 1] : 0.0f;
        v2f b[4];
#pragma unroll
        for (int t = 0; t < 4; ++t) {
            b[t].x = (k0     < K) ? Bc[t][(long)k0 * ldb]       : 0.0f;
            b[t].y = (k0 + 1 < K) ? Bc[t][(long)(k0 + 1) * ldb] : 0.0f;
        }
#pragma unroll
        for (int t = 0; t < 4; ++t)
            acc[t] = __builtin_amdgcn_wmma_f32_16x16x4_f32(false, a, false, b[t],
                                                           (short)0, acc[t], false, false);
    }

    // C/D layout: lane<16 -> M=j, N=sub ; lane>=16 -> M=j+8, N=sub
#pragma unroll
    for (int t = 0; t < 4; ++t) {
        const int colC = colbase0 + t * 16 + sub;
        if (colC < ncols) {
            const float bv = bias ? bias[colC] : 0.0f;
#pragma unroll
            for (int j = 0; j < 8; ++j) {
                const int r = rowbase + half * 8 + j;
                if (r < nrows) {
                    const float s = scaleA ? scaleA[r] : 1.0f;  // hoisted row scale
                    C[(long)r * ldc + colC] = acc[t][j] * s + bv;
                }
            }
        }
    }
}

// ---------------------------------------------------------------------------
// Edge scatter-add: agg[dst[e], :] += h[src[e], :]   (hid floats per edge)
// 128 threads per edge -> coalesced 512B per edge side, f32 atomics into L2
// ---------------------------------------------------------------------------
__global__ void scatter_add_kernel(const float* __restrict__ h,
                                   const int* __restrict__ src, const int* __restrict__ dst,
                                   float* __restrict__ agg, int E, int hid) {
    long t = (long)blockIdx.x * blockDim.x + threadIdx.x;
    long total = (long)E * hid;
    if (t < total) {
        int e = (int)(t / hid);
        int f = (int)(t - (long)e * hid);
        atomicAdd(&agg[(long)dst[e] * hid + f], h[(long)src[e] * hid + f]);
    }
}

// ---------------------------------------------------------------------------
// Post: out = relu(agg * rsqrt(deg_in)[row] + b[col]); optional fused residual:
//       out = relu(out + resid)
// ---------------------------------------------------------------------------
__global__ void post_kernel(const float* __restrict__ agg, const float* __restrict__ rs_in,
                            const float* __restrict__ b, const float* __restrict__ resid,
                            float* __restrict__ out, int N, int hid) {
    long t = (long)blockIdx.x * blockDim.x + threadIdx.x;
    long total = (long)N * hid;
    if (t < total) {
        int r = (int)(t / hid);
        int c = (int)(t - (long)r * hid);
        float v = fmaxf(agg[t] * rs_in[r] + b[c], 0.0f);
        if (resid) v = fmaxf(v + resid[t], 0.0f);
        out[t] = v;
    }
}

// ---------------------------------------------------------------------------
// Launch: 3 GCN layers + residual + fc head
// ---------------------------------------------------------------------------
extern "C" void kernel_launch(void* const* d_in, const int* in_sizes, int n_in,
                              void* d_out, int out_size, void* d_ws, size_t ws_size,
                              hipStream_t stream) {
    const float* features = (const float*)d_in[0];
    const int*   src      = (const int*)d_in[1];
    const int*   dst      = (const int*)d_in[2];
    const float* W1  = (const float*)d_in[3];
    const float* b1  = (const float*)d_in[4];
    const float* W2  = (const float*)d_in[5];
    const float* b2  = (const float*)d_in[6];
    const float* W3  = (const float*)d_in[7];
    const float* b3  = (const float*)d_in[8];
    const float* Wfc = (const float*)d_in[9];
    const float* bfc = (const float*)d_in[10];

    const int E    = in_sizes[1];
    const int HID  = in_sizes[4];              // 128
    const int INF  = in_sizes[3] / HID;        // 602
    const int OUTF = in_sizes[10];             // 41
    const int N    = in_sizes[0] / INF;        // 50000

    float* ws     = (float*)d_ws;
    float* rs_out = ws;                        // [N]  deg_out -> rsqrt
    float* rs_in  = ws + N;                    // [N]  deg_in  -> rsqrt
    float* hbuf   = ws + 2L * N;               // [N,HID]
    float* aggbuf = hbuf + (long)N * HID;      // [N,HID]
    float* x1buf  = aggbuf + (long)N * HID;    // [N,HID]
    float* x2buf  = x1buf + (long)N * HID;     // [N,HID]

    const int ZB = 1024;  // blocks for zero kernels (grid-stride)
    dim3 blk(256);
    // block tile = 64 rows x 128 cols (8 col tiles)
    dim3 gemm_grid((N + 63) / 64, ((HID + 15) / 16 + 7) / 8);   // grid.y == 1 for 128 cols
    dim3 fc_grid((N + 63) / 64, ((OUTF + 15) / 16 + 7) / 8);
    long scatN = (long)E * HID;
    dim3 scat_grid((unsigned)((scatN + 255) / 256));
    long postN = (long)N * HID;
    dim3 post_grid((unsigned)((postN + 255) / 256));

    // --- degrees -> rsqrt scales ---
    zero_kernel<<<ZB, blk, 0, stream>>>(rs_out, 2L * N);
    deg_kernel<<<(E + 255) / 256, blk, 0, stream>>>(src, dst, rs_out, rs_in, E);
    rsqrt_kernel<<<(2 * N + 255) / 256, blk, 0, stream>>>(rs_out, 2 * N);

    // --- layer 1: h = (feat * rs_out) @ W1 ; agg = scatter ; x1 = relu(agg*rs_in + b1) ---
    wmma_gemm_f32<<<gemm_grid, blk, 0, stream>>>(features, INF, rs_out, W1, HID, INF,
                                                 hbuf, HID, HID, nullptr, N);
    zero_kernel<<<ZB, blk, 0, stream>>>(aggbuf, (long)N * HID);
    scatter_add_kernel<<<scat_grid, blk, 0, stream>>>(hbuf, src, dst, aggbuf, E, HID);
    post_kernel<<<post_grid, blk, 0, stream>>>(aggbuf, rs_in, b1, nullptr, x1buf, N, HID);

    // --- layer 2 (reference computes it twice with identical inputs; _x == x) ---
    wmma_gemm_f32<<<gemm_grid, blk, 0, stream>>>(x1buf, HID, rs_out, W2, HID, HID,
                                                 hbuf, HID, HID, nullptr, N);
    zero_kernel<<<ZB, blk, 0, stream>>>(aggbuf, (long)N * HID);
    scatter_add_kernel<<<scat_grid, blk, 0, stream>>>(hbuf, src, dst, aggbuf, E, HID);
    post_kernel<<<post_grid, blk, 0, stream>>>(aggbuf, rs_in, b2, nullptr, x2buf, N, HID);

    // --- layer 3 + fused residual: y = relu(relu(agg*rs_in + b3) + x2) -> x1buf ---
    wmma_gemm_f32<<<gemm_grid, blk, 0, stream>>>(x2buf, HID, rs_out, W3, HID, HID,
                                                 hbuf, HID, HID, nullptr, N);
    zero_kernel<<<ZB, blk, 0, stream>>>(aggbuf, (long)N * HID);
    scatter_add_kernel<<<scat_grid, blk, 0, stream>>>(hbuf, src, dst, aggbuf, E, HID);
    post_kernel<<<post_grid, blk, 0, stream>>>(aggbuf, rs_in, b3, x2buf, x1buf, N, HID);

    // --- fc head: out = y @ Wfc + bfc  (column-predicated for OUTF=41) ---
    wmma_gemm_f32<<<fc_grid, blk, 0, stream>>>(x1buf, HID, nullptr, Wfc, OUTF, HID,
                                               (float*)d_out, OUTF, OUTF, bfc, N);
}